// SS2D_56564719288619
// MI455X (gfx1250) — compile-verified
//
#include <hip/hip_runtime.h>
#include <hip/hip_bf16.h>

// SS2D (VMamba selective-scan 2D) for gfx1250.
// Shapes: B=8, H=W=64, L=4096, D_MODEL=180, D_INNER=360, K=4 dirs, N=16 states, R=12.
// Workspace layout (floats), total ~355.5 MB:
//   z    [B,L,360]      11796480
//   xc   [B,L,360]      11796480   (reused as gated output 'g' after the scan)
//   xa   [B,L,360]      11796480
//   xdbl [B,4,L,48]      6291456   (dt_rank 12 | B 16 | C 16 | pad 4)
//   ybuf [B,4,L,360]    47185920

typedef float v2f __attribute__((ext_vector_type(2)));
typedef float v8f __attribute__((ext_vector_type(8)));

#define DM   180
#define DI   360
#define LL   4096
#define BSZ  8

// Spatial position (row-major h*64+w) that direction k's timestep t touches.
__device__ __forceinline__ int smap(int k, int t) {
    switch (k) {
        case 0:  return t;                                   // hw forward
        case 1:  return ((t & 63) << 6) | (t >> 6);          // wh forward (transpose)
        case 2:  return 4095 - t;                            // hw reverse
        default: { int u = 4095 - t; return ((u & 63) << 6) | (u >> 6); } // wh reverse
    }
}

// ---------------------------------------------------------------------------
// Kernel 1: in_proj GEMM  xz[m, 0:720] = x[m, 0:180] @ W^T, split into xc | z
// One wave per 16x16 output tile, fp32 WMMA 16x16x4. M=32768, N=720, K=180.
// ---------------------------------------------------------------------------
__global__ void k_inproj(const float* __restrict__ x, const float* __restrict__ w,
                         float* __restrict__ xc, float* __restrict__ z) {
    int wave = (blockIdx.x << 3) + (threadIdx.x >> 5);
    int lane = threadIdx.x & 31;
    int half = lane >> 4, mr = lane & 15;
    int mtile = wave / 45, ntile = wave - mtile * 45;

    const float* arow = x + (size_t)(mtile * 16 + mr) * DM;
    int col = ntile * 16 + mr;
    const float* bcol = w + (size_t)col * DM;

    v8f acc = {};
    for (int k = 0; k < DM; k += 4) {
        int ka = k + (half << 1);
        v2f a, b;
        a.x = arow[ka]; a.y = arow[ka + 1];
        b.x = bcol[ka]; b.y = bcol[ka + 1];
        acc = __builtin_amdgcn_wmma_f32_16x16x4_f32(false, a, false, b,
                                                    (short)0, acc, false, false);
    }
    int row0 = mtile * 16 + (half << 3);
#pragma unroll
    for (int j = 0; j < 8; ++j) {
        float v = acc[j];
        size_t m = (size_t)(row0 + j);
        if (col < DI) xc[m * DI + col] = v;
        else          z[m * DI + (col - DI)] = v;
    }
}

// ---------------------------------------------------------------------------
// Kernel 2: depthwise 3x3 SAME conv + bias + SiLU, NHWC layout.
// ---------------------------------------------------------------------------
__global__ void k_conv(const float* __restrict__ xc, const float* __restrict__ cw,
                       const float* __restrict__ cb, float* __restrict__ xa) {
    size_t id = (size_t)blockIdx.x * 256 + threadIdx.x;
    int d = (int)(id % DI);
    size_t p = id / DI;              // b*4096 + h*64 + w
    int w = (int)(p & 63);
    int h = (int)((p >> 6) & 63);
    int b = (int)(p >> 12);
    float acc = cb[d];
#pragma unroll
    for (int dy = 0; dy < 3; ++dy) {
        int hh = h + dy - 1;
        if (hh < 0 || hh > 63) continue;
#pragma unroll
        for (int dx = 0; dx < 3; ++dx) {
            int ww = w + dx - 1;
            if (ww < 0 || ww > 63) continue;
            acc += xc[((size_t)(b << 12) + (hh << 6) + ww) * DI + d] * cw[d * 9 + dy * 3 + dx];
        }
    }
    xa[id] = acc / (1.f + __expf(-acc));   // SiLU
}

// ---------------------------------------------------------------------------
// Kernel 3: x_proj GEMM per direction with gathered rows.
// xdbl[b,k,t,c] = sum_d xa[b, smap(k,t), d] * xpw[k,c,d].  N padded 44->48.
// ---------------------------------------------------------------------------
__global__ void k_xdbl(const float* __restrict__ xa, const float* __restrict__ xpw,
                       float* __restrict__ xdbl) {
    int wave = (blockIdx.x << 3) + (threadIdx.x >> 5);
    int lane = threadIdx.x & 31;
    int half = lane >> 4, mr = lane & 15;
    int k = wave / 6144;
    int rem = wave - k * 6144;
    int mtile = rem / 3, ntile = rem - mtile * 3;

    int m = mtile * 16 + mr;
    int b = m >> 12, t = m & 4095;
    const float* arow = xa + ((size_t)(b << 12) + smap(k, t)) * DI;
    int n = ntile * 16 + mr;
    const float* bcol = (n < 44) ? (xpw + ((size_t)k * 44 + n) * DI) : nullptr;

    v8f acc = {};
    for (int kk = 0; kk < DI; kk += 4) {
        int ka = kk + (half << 1);
        v2f a, bb;
        a.x = arow[ka]; a.y = arow[ka + 1];
        if (bcol) { bb.x = bcol[ka]; bb.y = bcol[ka + 1]; } else { bb.x = 0.f; bb.y = 0.f; }
        acc = __builtin_amdgcn_wmma_f32_16x16x4_f32(false, a, false, bb,
                                                    (short)0, acc, false, false);
    }
    int row0 = mtile * 16 + (half << 3);
#pragma unroll
    for (int j = 0; j < 8; ++j) {
        int mj = row0 + j;
        int bj = mj >> 12, tj = mj & 4095;
        xdbl[(((size_t)(bj * 4 + k) << 12) + tj) * 48 + n] = acc[j];
    }
}

// ---------------------------------------------------------------------------
// Kernel 4: selective scan. One single-wave block per 32-channel slice of each
// (b,k): grid = B*K*12 = 384 blocks, blockDim = 32.  Single-wave blocks spread
// the latency-bound recurrence across 12x more WGPs than block-per-(b,k), and
// allocate no barrier resources.  dt-proj (rank 12) fused; per-step dt/B/C
// vectors are wave-uniform float4 loads (SMEM-scalarizable).  Writes y at the
// spatial position so the merge is a plain deterministic 4-way sum.
// ---------------------------------------------------------------------------
__global__ void k_scan(const float* __restrict__ xa, const float* __restrict__ xdbl,
                       const float* __restrict__ dtw_all, const float* __restrict__ dtb,
                       const float* __restrict__ A_logs, const float* __restrict__ Ds,
                       float* __restrict__ ybuf) {
    int blk = blockIdx.x;          // (b*4 + k)*12 + slice
    int slice = blk % 12;
    int bk = blk / 12;             // b*4 + k
    int b = bk >> 2, k = bk & 3;
    int d = (slice << 5) + threadIdx.x;
    if (d >= DI) return;
    int kd = k * DI + d;

    float a[16], h[16], dtw[12];
#pragma unroll
    for (int n = 0; n < 16; ++n) { a[n] = -__expf(A_logs[(size_t)kd * 16 + n]); h[n] = 0.f; }
#pragma unroll
    for (int r = 0; r < 12; ++r) dtw[r] = dtw_all[(size_t)kd * 12 + r];
    float bias = dtb[kd];
    float Dd   = Ds[kd];

    const float* xbase = xa + ((size_t)b << 12) * DI;
    const float* drow  = xdbl + ((size_t)bk << 12) * 48;
    float*       yrow  = ybuf + ((size_t)bk << 12) * DI;

    for (int t = 0; t < LL; ++t) {
        float vals[44];
        const float4* q4 = reinterpret_cast<const float4*>(drow + (size_t)t * 48);
#pragma unroll
        for (int i = 0; i < 11; ++i) {
            float4 q = q4[i];
            vals[4 * i] = q.x; vals[4 * i + 1] = q.y; vals[4 * i + 2] = q.z; vals[4 * i + 3] = q.w;
        }
        if (t + 1 < LL) __builtin_prefetch(drow + (size_t)(t + 1) * 48, 0, 0);

        float dacc = bias;
#pragma unroll
        for (int r = 0; r < 12; ++r) dacc += dtw[r] * vals[r];
        float delta = (dacc > 20.f) ? dacc : log1pf(__expf(dacc));   // softplus

        int s = smap(k, t);
        float x_  = xbase[(size_t)s * DI + d];
        float dtx = delta * x_;
        float y   = Dd * x_;
#pragma unroll
        for (int n = 0; n < 16; ++n) {
            h[n] = __expf(delta * a[n]) * h[n] + dtx * vals[12 + n];
            y += h[n] * vals[28 + n];
        }
        yrow[(size_t)s * DI + d] = y;
    }
}

// ---------------------------------------------------------------------------
// Kernel 5: merge 4 directions + LayerNorm(360) + gamma/beta + SiLU(z) gate.
// One block per (b, spatial) pixel; wave32 shuffle + LDS reduction.
// ---------------------------------------------------------------------------
__global__ void k_merge(const float* __restrict__ ybuf, const float* __restrict__ z,
                        const float* __restrict__ gamma, const float* __restrict__ beta,
                        float* __restrict__ g) {
    int bs = blockIdx.x;               // b*4096 + s
    int b = bs >> 12, s = bs & 4095;
    int d = threadIdx.x;

    float y = 0.f;
    if (d < DI) {
#pragma unroll
        for (int k = 0; k < 4; ++k)
            y += ybuf[(((size_t)(b * 4 + k) << 12) + s) * DI + d];
    }
    float s1 = y, s2 = y * y;
#pragma unroll
    for (int off = 16; off > 0; off >>= 1) {
        s1 += __shfl_down(s1, off, 32);
        s2 += __shfl_down(s2, off, 32);
    }
    __shared__ float r1[12], r2[12];
    __shared__ float mu_s, rstd_s;
    int wv = threadIdx.x >> 5, ln = threadIdx.x & 31;
    if (ln == 0) { r1[wv] = s1; r2[wv] = s2; }
    __syncthreads();
    if (threadIdx.x == 0) {
        float t1 = 0.f, t2 = 0.f;
#pragma unroll
        for (int i = 0; i < 12; ++i) { t1 += r1[i]; t2 += r2[i]; }
        float mu = t1 * (1.f / 360.f);
        float var = t2 * (1.f / 360.f) - mu * mu;
        mu_s = mu;
        rstd_s = rsqrtf(var + 1e-5f);
    }
    __syncthreads();
    if (d < DI) {
        float zz  = z[(size_t)bs * DI + d];
        float sil = zz / (1.f + __expf(-zz));
        g[(size_t)bs * DI + d] = ((y - mu_s) * rstd_s * gamma[d] + beta[d]) * sil;
    }
}

// ---------------------------------------------------------------------------
// Kernel 6: out_proj GEMM  out[m, 0:180] = g[m, 0:360] @ W^T.  N padded to 192.
// ---------------------------------------------------------------------------
__global__ void k_outproj(const float* __restrict__ g, const float* __restrict__ w,
                          float* __restrict__ out) {
    int wave = (blockIdx.x << 3) + (threadIdx.x >> 5);
    int lane = threadIdx.x & 31;
    int half = lane >> 4, mr = lane & 15;
    int mtile = wave / 12, ntile = wave - mtile * 12;

    const float* arow = g + (size_t)(mtile * 16 + mr) * DI;
    int col = ntile * 16 + mr;
    const float* bcol = (col < DM) ? (w + (size_t)col * DI) : nullptr;

    v8f acc = {};
    for (int kk = 0; kk < DI; kk += 4) {
        int ka = kk + (half << 1);
        v2f a, bb;
        a.x = arow[ka]; a.y = arow[ka + 1];
        if (bcol) { bb.x = bcol[ka]; bb.y = bcol[ka + 1]; } else { bb.x = 0.f; bb.y = 0.f; }
        acc = __builtin_amdgcn_wmma_f32_16x16x4_f32(false, a, false, bb,
                                                    (short)0, acc, false, false);
    }
    if (col < DM) {
        int row0 = mtile * 16 + (half << 3);
#pragma unroll
        for (int j = 0; j < 8; ++j)
            out[(size_t)(row0 + j) * DM + col] = acc[j];
    }
}

// ---------------------------------------------------------------------------
extern "C" void kernel_launch(void* const* d_in, const int* in_sizes, int n_in,
                              void* d_out, int out_size, void* d_ws, size_t ws_size,
                              hipStream_t stream) {
    const float* x        = (const float*)d_in[0];
    const float* in_w     = (const float*)d_in[1];
    const float* conv_w   = (const float*)d_in[2];
    const float* conv_b   = (const float*)d_in[3];
    const float* xpw      = (const float*)d_in[4];
    const float* dtw      = (const float*)d_in[5];
    const float* dtb      = (const float*)d_in[6];
    const float* A_logs   = (const float*)d_in[7];
    const float* Ds       = (const float*)d_in[8];
    const float* gamma    = (const float*)d_in[9];
    const float* beta     = (const float*)d_in[10];
    const float* out_w    = (const float*)d_in[11];

    float* ws   = (float*)d_ws;
    float* z    = ws;                       // 11796480 floats
    float* xc   = z    + 11796480;          // 11796480
    float* xa   = xc   + 11796480;          // 11796480
    float* xdbl = xa   + 11796480;          //  6291456
    float* ybuf = xdbl + 6291456;           // 47185920
    float* g    = xc;                       // reuse xc after conv consumes it

    // in_proj: 2048 M-tiles x 45 N-tiles = 92160 waves -> 11520 blocks of 8 waves
    k_inproj<<<11520, 256, 0, stream>>>(x, in_w, xc, z);
    // depthwise conv + SiLU: 11796480 threads
    k_conv<<<46080, 256, 0, stream>>>(xc, conv_w, conv_b, xa);
    // x_proj per direction: 4 * 2048 * 3 = 24576 waves -> 3072 blocks
    k_xdbl<<<3072, 256, 0, stream>>>(xa, xpw, xdbl);
    // selective scan: single-wave blocks, 12 channel-slices per (b,k)
    k_scan<<<384, 32, 0, stream>>>(xa, xdbl, dtw, dtb, A_logs, Ds, ybuf);
    // merge + LN + gate: one block per pixel
    k_merge<<<32768, 384, 0, stream>>>(ybuf, z, gamma, beta, g);
    // out_proj: 2048 * 12 = 24576 waves -> 3072 blocks
    k_outproj<<<3072, 256, 0, stream>>>(g, out_w, (float*)d_out);
}